// Model_64063732187285
// MI455X (gfx1250) — compile-verified
//
#include <hip/hip_runtime.h>
#include <hip/hip_bf16.h>

#define CH       8
#define CONVS    8
#define HID      32
#define N_NODES  200000
#define N_EDGES  6400000
#define ENTRIES  2048
#define BN_EPS   1e-5f

typedef __attribute__((ext_vector_type(2))) float v2f;
typedef __attribute__((ext_vector_type(8))) float v8f;

// ---------------- degree / dinv ----------------
__global__ void k_init_deg(float* deg) {
    int i = blockIdx.x * blockDim.x + threadIdx.x;
    if (i < N_NODES) deg[i] = 1.0f;   // self loop
}
__global__ void k_deg_accum(const int* __restrict__ dst, float* __restrict__ deg) {
    int e = blockIdx.x * blockDim.x + threadIdx.x;
    if (e < N_EDGES) atomicAdd(&deg[dst[e]], 1.0f);
}
__global__ void k_dinv(float* deg) {
    int i = blockIdx.x * blockDim.x + threadIdx.x;
    if (i < N_NODES) deg[i] = rsqrtf(deg[i]);
}

// ---------------- embedding ----------------
__global__ void k_embed(const int* __restrict__ x, const float* __restrict__ emb,
                        float* __restrict__ h) {
    int idx = blockIdx.x * blockDim.x + threadIdx.x;
    if (idx >= N_NODES * CH) return;
    int i = idx >> 3, c = idx & 7;
    h[idx] = emb[x[i] * CH + c];
}

// ---------------- per-layer zero (agg + bn stats) ----------------
__global__ void k_zero_layer(float* __restrict__ agg, float* __restrict__ stats) {
    int idx = blockIdx.x * blockDim.x + threadIdx.x;
    if (idx < N_NODES * CH) agg[idx] = 0.0f;
    if (idx < 2 * CH) stats[idx] = 0.0f;
}

// ---------------- BatchNorm statistics: sum + sumsq per channel ----------------
__global__ void k_bn_stats(const float* __restrict__ h, float* __restrict__ stats) {
    __shared__ float ss[256][CH];
    __shared__ float sq[256][CH];
    int tid = threadIdx.x;
    float ls[CH], lq[CH];
#pragma unroll
    for (int c = 0; c < CH; ++c) { ls[c] = 0.0f; lq[c] = 0.0f; }
    for (int i = blockIdx.x * blockDim.x + tid; i < N_NODES; i += gridDim.x * blockDim.x) {
#pragma unroll
        for (int c = 0; c < CH; ++c) {
            float v = h[i * CH + c];
            ls[c] += v;
            lq[c] += v * v;
        }
    }
#pragma unroll
    for (int c = 0; c < CH; ++c) { ss[tid][c] = ls[c]; sq[tid][c] = lq[c]; }
    __syncthreads();
    for (int off = 128; off > 0; off >>= 1) {
        if (tid < off) {
#pragma unroll
            for (int c = 0; c < CH; ++c) {
                ss[tid][c] += ss[tid + off][c];
                sq[tid][c] += sq[tid + off][c];
            }
        }
        __syncthreads();
    }
    if (tid < CH)            atomicAdd(&stats[tid], ss[0][tid]);
    else if (tid < 2 * CH)   atomicAdd(&stats[tid], sq[0][tid - CH]);
}

// ---------------- BN + GCN linear transform via V_WMMA_F32_16X16X4_F32 ----------------
// One wave per 16-node tile.  ht[16x8] = hn[16x8] @ W[8x8]  (two K=4 WMMA steps).
__global__ void k_conv_transform(const float* __restrict__ h,
                                 float* __restrict__ ht,
                                 const float* __restrict__ stats,  // [0..7]=sum [8..15]=sumsq
                                 const float* __restrict__ gamma,  // [CONVS][CH]
                                 const float* __restrict__ beta,   // [CONVS][CH]
                                 const float* __restrict__ convw,  // [CONVS][CH][CH]
                                 int layer, int ntiles) {
    int wave = (blockIdx.x * blockDim.x + threadIdx.x) >> 5;
    int lane = threadIdx.x & 31;
    if (wave >= ntiles) return;                    // wave-uniform exit, EXEC stays all-1s
    int M  = lane & 15;                            // A row / B,D column
    int kh = lane >> 4;                            // K half-select
    int node = wave * 16 + M;

    const float* g  = gamma + layer * CH;
    const float* bt = beta  + layer * CH;
    const float* W  = convw + layer * CH * CH;

    int c00 = 2 * kh;                              // step-0 channels: c00, c00+1
    int c10 = 4 + 2 * kh;                          // step-1 channels: c10, c10+1

    float2 h0 = *(const float2*)(h + (size_t)node * CH + c00);
    float2 h1 = *(const float2*)(h + (size_t)node * CH + c10);

    auto bn = [&](float v, int c) {
        float mu  = stats[c] * (1.0f / N_NODES);
        float var = stats[CH + c] * (1.0f / N_NODES) - mu * mu;
        float rs  = rsqrtf(var + BN_EPS);
        return (v - mu) * rs * g[c] + bt[c];
    };
    v2f a0, a1;
    a0.x = bn(h0.x, c00); a0.y = bn(h0.y, c00 + 1);
    a1.x = bn(h1.x, c10); a1.y = bn(h1.y, c10 + 1);

    int n = M;                                     // output column
    v2f b0 = {0.0f, 0.0f}, b1 = {0.0f, 0.0f};
    if (n < CH) {                                  // cols 8..15 are zero padding
        b0.x = W[c00 * CH + n];       b0.y = W[(c00 + 1) * CH + n];
        b1.x = W[c10 * CH + n];       b1.y = W[(c10 + 1) * CH + n];
    }
    v8f acc = {};
    acc = __builtin_amdgcn_wmma_f32_16x16x4_f32(false, a0, false, b0, (short)0, acc, false, false);
    acc = __builtin_amdgcn_wmma_f32_16x16x4_f32(false, a1, false, b1, (short)0, acc, false, false);

    if (n < CH) {
        int rowbase = wave * 16 + 8 * kh;          // D vgpr r -> row r + 8*kh
#pragma unroll
        for (int r = 0; r < 8; ++r)
            ht[(size_t)(rowbase + r) * CH + n] = acc[r];
    }
}

// ---------------- edge aggregation: agg[dst] += ht[src] * dinv[src]*dinv[dst] ----------------
__global__ void k_edge_agg(const int* __restrict__ src, const int* __restrict__ dst,
                           const float* __restrict__ dinv, const float* __restrict__ ht,
                           float* __restrict__ agg) {
    long long idx = (long long)blockIdx.x * blockDim.x + threadIdx.x;
    if (idx >= (long long)N_EDGES * CH) return;
    int e = (int)(idx >> 3);
    int c = (int)(idx & 7);
    int s = src[e], d = dst[e];
    float w = dinv[s] * dinv[d];
    atomicAdd(&agg[(size_t)d * CH + c], ht[(size_t)s * CH + c] * w);
}

// ---------------- residual + self-loop + bias + ReLU ----------------
__global__ void k_finalize(float* __restrict__ h, const float* __restrict__ agg,
                           const float* __restrict__ ht, const float* __restrict__ dinv,
                           const float* __restrict__ convb, int layer) {
    int idx = blockIdx.x * blockDim.x + threadIdx.x;
    if (idx >= N_NODES * CH) return;
    int i = idx >> 3, c = idx & 7;
    float sn = dinv[i] * dinv[i];
    float v = h[idx] + agg[idx] + ht[idx] * sn + convb[layer * CH + c];
    h[idx] = fmaxf(v, 0.0f);
}

// ---------------- entry gather + projection via WMMA ----------------
// AB[i][0:32]  = xe[i] @ hidden_w[0:8]    (A part)
// AB[i][32:64] = xe[i] @ hidden_w[8:16]   (B part)
__global__ void k_entry_proj(const float* __restrict__ h,
                             const int* __restrict__ entry_index,
                             const float* __restrict__ hw,   // [2*CH][HID]
                             float* __restrict__ AB) {       // [ENTRIES][2*HID]
    int wave = (blockIdx.x * blockDim.x + threadIdx.x) >> 5;
    int lane = threadIdx.x & 31;
    const int ntiles = ENTRIES / 16;
    if (wave >= ntiles) return;
    int M  = lane & 15;
    int kh = lane >> 4;
    int row  = wave * 16 + M;
    int node = entry_index[row];
    int c00 = 2 * kh, c10 = 4 + 2 * kh;

    float2 x0 = *(const float2*)(h + (size_t)node * CH + c00);
    float2 x1 = *(const float2*)(h + (size_t)node * CH + c10);
    v2f a0; a0.x = x0.x; a0.y = x0.y;
    v2f a1; a1.x = x1.x; a1.y = x1.y;

    int n = M;
#pragma unroll
    for (int nt = 0; nt < 4; ++nt) {
        int j = nt * 16 + n;                        // 0..63, j<32 uniform per nt
        int roff = (j < HID) ? 0 : CH;              // top rows for A, bottom rows for B
        int jj   = (j < HID) ? j : (j - HID);
        v2f b0, b1;
        b0.x = hw[(roff + c00) * HID + jj];
        b0.y = hw[(roff + c00 + 1) * HID + jj];
        b1.x = hw[(roff + c10) * HID + jj];
        b1.y = hw[(roff + c10 + 1) * HID + jj];
        v8f acc = {};
        acc = __builtin_amdgcn_wmma_f32_16x16x4_f32(false, a0, false, b0, (short)0, acc, false, false);
        acc = __builtin_amdgcn_wmma_f32_16x16x4_f32(false, a1, false, b1, (short)0, acc, false, false);
        int rowbase = wave * 16 + 8 * kh;
#pragma unroll
        for (int r = 0; r < 8; ++r)
            AB[(size_t)(rowbase + r) * (2 * HID) + j] = acc[r];
    }
}

// ---------------- pairwise: out[i,j] = sum_k relu(A[i,k]+B[j,k]+hb[k])*wo[k] + ob ----------------
__global__ void k_pairwise(const float* __restrict__ AB,
                           const float* __restrict__ hb,
                           const float* __restrict__ wo,
                           const float* __restrict__ ob,
                           float* __restrict__ out) {
    __shared__ float sA[16][HID];
    __shared__ float sB[16][HID];
    __shared__ float shb[HID];
    __shared__ float swo[HID];
    int tid = threadIdx.x;
    int bi = blockIdx.x, bj = blockIdx.y;
    for (int t = tid; t < 16 * HID; t += 256) {
        int r = t / HID, c = t % HID;
        sA[r][c] = AB[(size_t)(bi * 16 + r) * (2 * HID) + c];
        sB[r][c] = AB[(size_t)(bj * 16 + r) * (2 * HID) + HID + c];
    }
    if (tid < HID) { shb[tid] = hb[tid]; swo[tid] = wo[tid]; }
    __syncthreads();
    int j = tid & 15;
    int i = tid >> 4;
    float acc = ob[0];
#pragma unroll
    for (int k = 0; k < HID; ++k) {
        float v = sA[i][k] + sB[j][k] + shb[k];
        acc = fmaf(fmaxf(v, 0.0f), swo[k], acc);
    }
    out[(size_t)(bi * 16 + i) * ENTRIES + (bj * 16 + j)] = acc;
}

extern "C" void kernel_launch(void* const* d_in, const int* in_sizes, int n_in,
                              void* d_out, int out_size, void* d_ws, size_t ws_size,
                              hipStream_t stream) {
    const int*   x    = (const int*)d_in[0];
    const int*   ei   = (const int*)d_in[1];   // [2][E]
    const int*   eidx = (const int*)d_in[2];
    const float* emb  = (const float*)d_in[3];
    const float* gma  = (const float*)d_in[4];
    const float* bta  = (const float*)d_in[5];
    const float* cw   = (const float*)d_in[6];
    const float* cb   = (const float*)d_in[7];
    const float* hw   = (const float*)d_in[8];
    const float* hb   = (const float*)d_in[9];
    const float* wo   = (const float*)d_in[10];
    const float* ob   = (const float*)d_in[11];
    float* out = (float*)d_out;

    float* ws    = (float*)d_ws;
    float* dinv  = ws;                              // N
    float* h     = ws + (size_t)N_NODES;            // N*8
    float* ht    = ws + (size_t)N_NODES * 9;        // N*8
    float* agg   = ws + (size_t)N_NODES * 17;       // N*8
    float* stats = ws + (size_t)N_NODES * 25;       // 16
    float* AB    = stats + 16;                      // ENTRIES*64

    const int* src = ei;
    const int* dst = ei + N_EDGES;

    dim3 b256(256);
    k_init_deg <<<(N_NODES + 255) / 256, b256, 0, stream>>>(dinv);
    k_deg_accum<<<(N_EDGES + 255) / 256, b256, 0, stream>>>(dst, dinv);
    k_dinv     <<<(N_NODES + 255) / 256, b256, 0, stream>>>(dinv);
    k_embed    <<<(N_NODES * CH + 255) / 256, b256, 0, stream>>>(x, emb, h);

    const int ntiles = N_NODES / 16;                // 12500, exact
    const long long ec = (long long)N_EDGES * CH;   // 51.2M
    for (int l = 0; l < CONVS; ++l) {
        k_zero_layer    <<<(N_NODES * CH + 255) / 256, b256, 0, stream>>>(agg, stats);
        k_bn_stats      <<<256, b256, 0, stream>>>(h, stats);
        k_conv_transform<<<(ntiles * 32 + 255) / 256, b256, 0, stream>>>(h, ht, stats, gma, bta, cw, l, ntiles);
        k_edge_agg      <<<(unsigned)((ec + 255) / 256), b256, 0, stream>>>(src, dst, dinv, ht, agg);
        k_finalize      <<<(N_NODES * CH + 255) / 256, b256, 0, stream>>>(h, agg, ht, dinv, cb, l);
    }

    k_entry_proj<<<((ENTRIES / 16) * 32 + 255) / 256, b256, 0, stream>>>(h, eidx, hw, AB);
    dim3 pg(ENTRIES / 16, ENTRIES / 16);
    k_pairwise<<<pg, b256, 0, stream>>>(AB, hb, wo, ob, out);
}